// casConv2d_63505386438716
// MI455X (gfx1250) — compile-verified
//
#include <hip/hip_runtime.h>

typedef __attribute__((ext_vector_type(2))) float v2f;
typedef __attribute__((ext_vector_type(8))) float v8f;

#define C_IN   128
#define H_IN   32
#define W_IN   32
#define OC_N   128
#define CKK    1152        // C_IN * 3 * 3
#define L_N    1024        // 32 * 32 output locations
#define NCHUNK 36          // 1152 / 32 (front zero-pad chunk contributes 0 -> skipped)
#define QMAXF  255.0f

// ---------------- Phase 0: materialize unfolded input cols[t][l] --------------------
// t = c*9 + kh*3 + kw (channel-major), l = oh*32 + ow, zero padding at borders.
__global__ void casconv_unfold_kernel(const float* __restrict__ x,
                                      float* __restrict__ cols) {
    const int idx = blockIdx.x * blockDim.x + threadIdx.x;  // t*L_N + l, l fastest
    if (idx >= CKK * L_N) return;
    const int t = idx >> 10;
    const int l = idx & (L_N - 1);
    const int c  = t / 9;
    const int r  = t - c * 9;
    const int kh = r / 3;
    const int kw = r - kh * 3;
    const int ih = (l >> 5) + kh - 1;   // PAD = 1
    const int iw = (l & 31) + kw - 1;
    float v = 0.0f;
    if ((unsigned)ih < (unsigned)H_IN && (unsigned)iw < (unsigned)W_IN)
        v = x[(c * H_IN + ih) * W_IN + iw];
    cols[idx] = v;
}

// ---------------- Phase A: full conv (+bias) via f32 WMMA, no quantization ----------
__global__ void casconv_total_kernel(const float* __restrict__ cols,
                                     const float* __restrict__ w,
                                     const float* __restrict__ bias,
                                     float* __restrict__ total) {
    const int lane   = threadIdx.x & 31;
    const int wave   = threadIdx.x >> 5;
    const int wglob  = blockIdx.x * 8 + wave;   // 512 waves total
    const int m_tile = wglob & 7;               // 8 tiles over OC=128
    const int n_tile = wglob >> 3;              // 64 tiles over L=1024
    const int m  = lane & 15;
    const int g  = lane >> 4;                   // half-wave group: K = vgpr + 2*g
    const int oc = m_tile * 16 + m;
    const int l  = n_tile * 16 + m;

    const float* ap = w    + oc * CKK   + 2 * g;   // A(oc, t..t+1)
    const float* bp = cols + 2 * g * L_N + l;      // B(t..t+1, l)

    v8f acc = {};
    for (int j = 0; j < NCHUNK; ++j) {
#pragma unroll
        for (int kk = 0; kk < 8; ++kk) {
            v2f a = *(const v2f*)(ap + kk * 4);            // aligned b64
            v2f b;
            b.x = bp[(kk * 4)     * L_N];                  // immediate offsets
            b.y = bp[(kk * 4 + 1) * L_N];
            acc = __builtin_amdgcn_wmma_f32_16x16x4_f32(
                false, a, false, b, (short)0, acc, false, false);
        }
        ap += 32;
        bp += 32 * L_N;
    }
#pragma unroll
    for (int r = 0; r < 8; ++r) {
        const int row = m_tile * 16 + r + 8 * g;   // C/D layout: M = r + 8*g
        total[row * L_N + l] = acc[r] + bias[row];
    }
}

// ---------------- Phase B: per-location quant params (min/max over OC) --------------
__global__ void casconv_qparam_kernel(const float* __restrict__ total,
                                      float* __restrict__ scale,
                                      float* __restrict__ zp) {
    const int l = blockIdx.x * blockDim.x + threadIdx.x;
    if (l >= L_N) return;
    float mx = -__builtin_inff();
    float mn =  __builtin_inff();
    for (int oc = 0; oc < OC_N; ++oc) {           // coalesced: contiguous in l
        float v = total[oc * L_N + l];
        mx = fmaxf(mx, v);
        mn = fminf(mn, v);
    }
    float s = (mx - mn) / QMAXF;
    float z = -mn / s;
    if (z != z) z = 0.0f;                          // NaN -> 0 (matches jnp.where(isnan))
    z = fminf(fmaxf(z, 0.0f), QMAXF);
    z = truncf(z);                                 // .int() truncation
    scale[l] = s;
    zp[l]    = z;
}

// ---------------- Phase C: per-chunk WMMA partial sums + fake-quant + reduce --------
__global__ void casconv_quant_kernel(const float* __restrict__ cols,
                                     const float* __restrict__ w,
                                     const float* __restrict__ scale,
                                     const float* __restrict__ zp,
                                     float* __restrict__ out) {
    const int lane   = threadIdx.x & 31;
    const int wave   = threadIdx.x >> 5;
    const int wglob  = blockIdx.x * 8 + wave;
    const int m_tile = wglob & 7;
    const int n_tile = wglob >> 3;
    const int m  = lane & 15;
    const int g  = lane >> 4;
    const int oc = m_tile * 16 + m;
    const int l  = n_tile * 16 + m;

    const float s = scale[l];                      // every lane owns one column
    const float z = zp[l];

    const float* ap = w    + oc * CKK   + 2 * g;
    const float* bp = cols + 2 * g * L_N + l;

    v8f result = {};
    for (int j = 0; j < NCHUNK; ++j) {
        v8f cacc = {};                             // fresh accumulator per chunk
#pragma unroll
        for (int kk = 0; kk < 8; ++kk) {
            v2f a = *(const v2f*)(ap + kk * 4);
            v2f b;
            b.x = bp[(kk * 4)     * L_N];
            b.y = bp[(kk * 4 + 1) * L_N];
            cacc = __builtin_amdgcn_wmma_f32_16x16x4_f32(
                false, a, false, b, (short)0, cacc, false, false);
        }
        ap += 32;
        bp += 32 * L_N;
#pragma unroll
        for (int r = 0; r < 8; ++r) {
            float q = __builtin_rintf(cacc[r] / s) + z;   // RNE matches jnp.round
            q = fminf(fmaxf(q, 0.0f), QMAXF);
            result[r] += (q - z) * s;                     // dequant (forward value)
        }
    }
#pragma unroll
    for (int r = 0; r < 8; ++r) {
        const int row = m_tile * 16 + r + 8 * g;
        out[row * L_N + l] = result[r];
    }
}

extern "C" void kernel_launch(void* const* d_in, const int* in_sizes, int n_in,
                              void* d_out, int out_size, void* d_ws, size_t ws_size,
                              hipStream_t stream) {
    (void)in_sizes; (void)n_in; (void)out_size; (void)ws_size;
    const float* x    = (const float*)d_in[0];   // (1,128,32,32)
    const float* wgt  = (const float*)d_in[1];   // (128,128,3,3) == (OC, CKK)
    const float* bias = (const float*)d_in[2];   // (128,)
    float* out = (float*)d_out;                  // (1,128,32,32) f32

    // workspace: cols (1152*1024) | total (128*1024) | scale (1024) | zp (1024) f32
    float* cols  = (float*)d_ws;
    float* total = cols + CKK * L_N;
    float* scale = total + OC_N * L_N;
    float* zp    = scale + L_N;

    casconv_unfold_kernel<<<(CKK * L_N + 255) / 256, 256, 0, stream>>>(x, cols);
    // 512 waves = 8 (OC tiles) x 64 (L tiles); 8 waves per 256-thread block
    casconv_total_kernel<<<64, 256, 0, stream>>>(cols, wgt, bias, total);
    casconv_qparam_kernel<<<4, 256, 0, stream>>>(total, scale, zp);
    casconv_quant_kernel<<<64, 256, 0, stream>>>(cols, wgt, scale, zp, out);
}